// EmbeddingShard_8383776162316
// MI455X (gfx1250) — compile-verified
//
#include <hip/hip_runtime.h>

#define N_VOCAB 50400
#define D_MODEL 4096
#define SHARDS  8
#define SEQ     2048
#define BATCH   2
#define PER_IN  (N_VOCAB / SHARDS)   // 6300

typedef __attribute__((ext_vector_type(4))) unsigned int u32x4;
typedef __attribute__((ext_vector_type(8))) int          i32x8;
typedef __attribute__((ext_vector_type(4))) int          i32x4;
typedef __attribute__((ext_vector_type(4))) float        f32x4;

#if defined(__has_builtin)
#  if __has_builtin(__builtin_amdgcn_tensor_load_to_lds) && \
      __has_builtin(__builtin_amdgcn_s_wait_tensorcnt)
#    define USE_TDM 1
#  endif
#endif
#ifndef USE_TDM
#  define USE_TDM 0
#endif

// ---------------------------------------------------------------------------
// Kernel 1: precompute per-owner boundary sums  B[o][d] =
//   sum_{s>o} kernel[6300*s, d] + sum_{s<o} kernel[6300*s+6299, d] + 8*bias[d]
// The 16 boundary rows are all the 7 "clipped" shard contributions can touch.
// ---------------------------------------------------------------------------
__global__ __launch_bounds__(256) void boundary_precompute(
    const float* __restrict__ kern, const float* __restrict__ bias,
    float* __restrict__ B) {
  const int d = blockIdx.x * blockDim.x + threadIdx.x;
  if (d >= D_MODEL) return;
  float lo[SHARDS], hi[SHARDS];
#pragma unroll
  for (int s = 0; s < SHARDS; ++s) {
    lo[s] = kern[(size_t)(s * PER_IN) * D_MODEL + d];
    hi[s] = kern[(size_t)(s * PER_IN + PER_IN - 1) * D_MODEL + d];
  }
  const float bb = 8.0f * bias[d];
#pragma unroll
  for (int o = 0; o < SHARDS; ++o) {
    float acc = bb;
    for (int s = o + 1; s < SHARDS; ++s) acc += lo[s];
    for (int s = 0; s < o; ++s)         acc += hi[s];
    B[o * D_MODEL + d] = acc;
  }
}

// ---------------------------------------------------------------------------
// Kernel 2: one workgroup (256 threads = 8 wave32) per token.
//   out[token, :] = kernel[x, :] + B[owner, :] + pos_flat[t*4096 : ...]
// Wave 0 kicks a TDM DMA of the 16KB pos row into LDS; ALL waves then issue
// their gather loads (kernel row NT, B row RT) so the DMA and the VGPR loads
// are in flight together; only then do we wait TENSORcnt + barrier, read the
// staged tile from LDS, and stream the result out with NT stores.
// ---------------------------------------------------------------------------
#define ITERS ((D_MODEL / 4) / 256)   // 4 float4 per thread

__global__ __launch_bounds__(256) void embed_gather(
    const int* __restrict__ tok, const float* __restrict__ kern,
    const float* __restrict__ pos, const float* __restrict__ B,
    float* __restrict__ out) {
  __shared__ float lds_pos[D_MODEL];          // 16 KB of the 320 KB WGP LDS

  const int token = blockIdx.x;               // [0, BATCH*SEQ)
  const int t     = token & (SEQ - 1);
  const int xi    = tok[token];
  const int owner = xi / PER_IN;

  const f32x4* __restrict__ krow = (const f32x4*)(kern + (size_t)xi * D_MODEL);
  const f32x4* __restrict__ brow = (const f32x4*)(B + (size_t)owner * D_MODEL);
  const float* __restrict__ prow = pos + (size_t)t * D_MODEL;
  f32x4*       __restrict__ orow = (f32x4*)(out + (size_t)token * D_MODEL);

#if USE_TDM
  if ((threadIdx.x >> 5) == 0) {              // wave 0 only issues the DMA
    // LDS byte offset = low 32 bits of the flat shared-aperture address.
    const unsigned lds_off = (unsigned)(size_t)(const void*)lds_pos;
    const unsigned long long gaddr = (unsigned long long)(const void*)prow;

    // D# group 0: count=1, lds_addr, global_addr[56:0], type=2 ("image")
    u32x4 g0 = {1u,
                lds_off,
                (unsigned)(gaddr & 0xFFFFFFFFull),
                (((unsigned)(gaddr >> 32)) & 0x01FFFFFFu) | (2u << 30)};
    // D# group 1: wg_mask=0, data_size=2 (4B); 1D tile: tensor_dim0=4096,
    // tensor_dim1=1, tile_dim0=4096, tile_dim1=1, tensor_dim0_stride=4096.
    i32x8 g1 = {(int)0x00020000,              // data_size=2, no barrier/pad/iter
                (int)((4096u & 0xFFFFu) << 16),   // tensor_dim0[15:0] @ [63:48]
                (int)(1u << 16),              // tensor_dim0[31:16]=0, tensor_dim1=1
                (int)(4096u << 16),           // tensor_dim1 hi=0, tile_dim0=4096
                1,                            // tile_dim1=1, tile_dim2=0
                4096,                         // tensor_dim0_stride[31:0]
                0, 0};                        // stride hi / dim1_stride = 0
    i32x4 z4 = {0, 0, 0, 0};                  // groups 2/3 unused (<=2D tensor)
#if __clang_major__ >= 23
    i32x8 z8 = {0, 0, 0, 0, 0, 0, 0, 0};
    __builtin_amdgcn_tensor_load_to_lds(g0, g1, z4, z4, z8, 0);
#else
    __builtin_amdgcn_tensor_load_to_lds(g0, g1, z4, z4, 0);
#endif
  }
#endif

  // Issue all gather loads while the TDM transfer is in flight.
  f32x4 k4[ITERS], b4[ITERS];
#pragma unroll
  for (int i = 0; i < ITERS; ++i) {
    const int idx = threadIdx.x + i * 256;
    k4[i] = __builtin_nontemporal_load(&krow[idx]);  // read-once row: NT
    b4[i] = brow[idx];                               // keep B L2-hot: RT
  }

#if USE_TDM
  if ((threadIdx.x >> 5) == 0) {
    __builtin_amdgcn_s_wait_tensorcnt(0);     // TENSORcnt==0: tile is in LDS
  }
  __syncthreads();
  const f32x4* __restrict__ prow4 = (const f32x4*)lds_pos;
#else
  const f32x4* __restrict__ prow4 = (const f32x4*)prow;
#endif

#pragma unroll
  for (int i = 0; i < ITERS; ++i) {
    const int idx = threadIdx.x + i * 256;
    f32x4 r = k4[i] + b4[i] + prow4[idx];
    __builtin_nontemporal_store(r, &orow[idx]);      // write-once: NT store
  }
}

extern "C" void kernel_launch(void* const* d_in, const int* in_sizes, int n_in,
                              void* d_out, int out_size, void* d_ws, size_t ws_size,
                              hipStream_t stream) {
  const int*   x    = (const int*)d_in[0];    // [BATCH, SEQ] int32
  const float* kern = (const float*)d_in[1];  // [N_VOCAB, D_MODEL] f32
  const float* bias = (const float*)d_in[2];  // [D_MODEL] f32
  const float* pos  = (const float*)d_in[3];  // [SHARDS, SEQ, PER_OUT] f32 (flat == [SEQ, D_MODEL])
  float*       out  = (float*)d_out;          // [BATCH, SEQ, D_MODEL] f32
  float*       B    = (float*)d_ws;           // 8 * 4096 * 4 = 128 KB scratch

  boundary_precompute<<<D_MODEL / 256, 256, 0, stream>>>(kern, bias, B);
  embed_gather<<<BATCH * SEQ, 256, 0, stream>>>(x, kern, pos, B, out);
}